// Pred3DPose_42399917146366
// MI455X (gfx1250) — compile-verified
//
#include <hip/hip_runtime.h>
#include <stdint.h>
#include <stddef.h>

// ---------------------------------------------------------------------------
// CDNA5 (gfx1250, wave32) implementation of the Pred3DPose pipeline.
// All GEMMs run through V_WMMA_F32_16X16X4_F32 (full fp32 precision).
// B panels staged pair-interleaved in LDS (contiguous b64 fragments);
// dense-K layers use b64 A loads with zero bookkeeping, strided/transposed
// layers use division-free incremental offset mapping.
// ---------------------------------------------------------------------------

typedef float v2f __attribute__((ext_vector_type(2)));
typedef float v8f __attribute__((ext_vector_type(8)));

#define GEMM_WAVES 4
#define KP_MAX 160   // max padded K over all layers (pred0: 152)

// Generic strided A-matrix address map:
//   addr(m,k) = (m/ic)*os + (m%ic)*is + (k/kc)*kos + (k%kc)*ks
struct AMap { int ic, os, is, kc, kos, ks; };

__device__ __forceinline__ v8f wmma4(v2f a, v2f b, v8f c) {
  return __builtin_amdgcn_wmma_f32_16x16x4_f32(
      false, a, false, b, (short)0, c, false, false);
}

// ---------------------------------------------------------------------------
// GEMM: Out[M x N] = A[M x Ka] * Wp[Kp x Np] + bias[N]
// Wp zero-padded (Kp % 4 == 0, Np % 16 == 0). One wave owns one (TWO: two)
// 16x16 C tile(s). Fragment layouts per CDNA5 ISA 7.12.2:
//   A 16x4 f32: lanes 0-15 -> M=lane, v[0]=K0 v[1]=K1; lanes 16-31 -> K2,K3
//   B 4x16 f32: lanes 0-15 -> N=lane, v[0]=K0 v[1]=K1; lanes 16-31 -> K2,K3
//   C 16x16:    v[i] -> row m0+i (+8 for lanes 16-31), col = lane&15
// LDS B layout is pair-interleaved: bsh[((k>>1)*LDB + c)*2 + (k&1)], so a
// (K,K+1) fragment for one column is a single aligned 8-byte ds load.
// ---------------------------------------------------------------------------
template <bool TWO, bool DENSE>
__global__ void gemm_wmma_f32(const float* __restrict__ A,
                              const float* __restrict__ Wp,
                              const float* __restrict__ bias,
                              float* __restrict__ Out,
                              int M, int N, int Np, int Ka, int Kp,
                              AMap am)
{
  constexpr int LDB = TWO ? 32 : 16;
  __shared__ float bsh[KP_MAX * LDB];

  const int tid = threadIdx.x;
  const int n0  = blockIdx.x * LDB;

  // ---- cooperative stage of the B panel into LDS (pair-interleaved) ----
  for (int i = tid; i < Kp * LDB; i += 128) {
    const int k  = i / LDB;                 // LDB is a power of two
    const int c  = i & (LDB - 1);
    const int gc = n0 + c;
    const float v = (gc < Np) ? Wp[(size_t)k * Np + gc] : 0.f;
    bsh[(((k >> 1) * LDB + c) << 1) + (k & 1)] = v;
  }
  __syncthreads();

  const int lane = tid & 31;
  const int wv   = tid >> 5;
  const int m0   = (blockIdx.y * GEMM_WAVES + wv) << 4;
  if (m0 >= M) return;                      // wave-uniform exit

  const int lg = lane >> 4;                 // half-wave group (0/1)
  const int ml = lane & 15;

  // Row clamped into range: rows >= M produce garbage outputs that are
  // never consumed (all A-maps address only valid rows).
  int row = m0 + ml; if (row >= M) row = M - 1;

  v8f acc0 = {0.f,0.f,0.f,0.f,0.f,0.f,0.f,0.f};
  v8f acc1 = {0.f,0.f,0.f,0.f,0.f,0.f,0.f,0.f};

  int ib = lg * LDB + ml;                   // B pair index for (ka>>1, col)
  const int Kmain = Kp - 4;                 // peel the final iteration

  if (DENSE) {
    // Dense rows: A[row*is + k], is == Ka (even) -> aligned b64 A fragments.
    const float* Ap = A + (size_t)row * am.is;
    const int la = lg << 1;
    int k0 = 0;
    for (; k0 < Kmain; k0 += 4) {
      v2f a  = *(const v2f*)(Ap + k0 + la);
      v2f b  = *(const v2f*)(bsh + (ib << 1));
      acc0 = wmma4(a, b, acc0);
      if (TWO) {
        v2f b2 = *(const v2f*)(bsh + ((ib + 16) << 1));
        acc1 = wmma4(a, b2, acc1);
      }
      ib += 2 * LDB;
    }
    // peeled last iteration: pair-granular validity (Ka even)
    {
      const int ka0 = k0 + la;
      const bool pv = (ka0 + 1 < Ka);
      v2f t = *(const v2f*)(Ap + (pv ? ka0 : 0));
      v2f a; a.x = pv ? t.x : 0.f; a.y = pv ? t.y : 0.f;
      v2f b = *(const v2f*)(bsh + (ib << 1));
      acc0 = wmma4(a, b, acc0);
      if (TWO) {
        v2f b2 = *(const v2f*)(bsh + ((ib + 16) << 1));
        acc1 = wmma4(a, b2, acc1);
      }
    }
  } else {
    // Mapped rows: incremental (k/kc, k%kc) offsets, branch-free rollover.
    const float* Ar = A + (row / am.ic) * am.os + (row % am.ic) * am.is;
    int r0 = (lg << 1), r1 = r0 + 1;        // k in {0..3} < kc always
    int off0 = r0 * am.ks, off1 = r1 * am.ks;
    const int stepo = am.ks << 2;
    const int adj   = am.kos - am.kc * am.ks;

    for (int k0 = 0; k0 < Kmain; k0 += 4) {
      v2f a, b;
      a.x = Ar[off0];
      a.y = Ar[off1];
      b = *(const v2f*)(bsh + (ib << 1));
      acc0 = wmma4(a, b, acc0);
      if (TWO) {
        v2f b2 = *(const v2f*)(bsh + ((ib + 16) << 1));
        acc1 = wmma4(a, b2, acc1);
      }
      r0 += 4; off0 += stepo;
      r1 += 4; off1 += stepo;
      const bool c0 = (r0 >= am.kc); r0 -= c0 ? am.kc : 0; off0 += c0 ? adj : 0;
      const bool c1 = (r1 >= am.kc); r1 -= c1 ? am.kc : 0; off1 += c1 ? adj : 0;
      ib += 2 * LDB;
    }
    // peeled last iteration: only place ka may reach Ka (Kp = rup4(Ka))
    {
      const int ka0 = Kmain + (lg << 1);
      const bool v0 = (ka0     < Ka);
      const bool v1 = (ka0 + 1 < Ka);
      v2f a, b;
      float t0 = Ar[v0 ? off0 : 0];
      float t1 = Ar[v1 ? off1 : 0];
      a.x = v0 ? t0 : 0.f;
      a.y = v1 ? t1 : 0.f;
      b = *(const v2f*)(bsh + (ib << 1));
      acc0 = wmma4(a, b, acc0);
      if (TWO) {
        v2f b2 = *(const v2f*)(bsh + ((ib + 16) << 1));
        acc1 = wmma4(a, b2, acc1);
      }
    }
  }

  // ---- epilogue: bias add + guarded store (Out is compact M x N) ----
  const int col0 = n0 + ml;
  if (col0 < N) {
    const float bs = bias ? bias[col0] : 0.f;
#pragma unroll
    for (int i = 0; i < 8; ++i) {
      const int r = m0 + i + (lg << 3);
      if (r < M) Out[(size_t)r * N + col0] = acc0[i] + bs;
    }
  }
  if (TWO) {
    const int col1 = n0 + 16 + ml;
    if (col1 < N) {
      const float bs = bias ? bias[col1] : 0.f;
#pragma unroll
      for (int i = 0; i < 8; ++i) {
        const int r = m0 + i + (lg << 3);
        if (r < M) Out[(size_t)r * N + col1] = acc1[i] + bs;
      }
    }
  }
}

// ---------------------------------------------------------------------------
// LayerNorm (+ReLU) in place over the last dim N of X[rows x N].
// ---------------------------------------------------------------------------
__global__ void lnrelu_f32(float* __restrict__ X,
                           const float* __restrict__ g,
                           const float* __restrict__ b,
                           int rows, int N)
{
  const int r = blockIdx.x * 4 + (threadIdx.x >> 5);
  if (r >= rows) return;
  const int lane = threadIdx.x & 31;
  float* xr = X + (size_t)r * N;

  float s = 0.f, s2 = 0.f;
  for (int j = lane; j < N; j += 32) { float v = xr[j]; s += v; s2 += v * v; }
#pragma unroll
  for (int off = 16; off > 0; off >>= 1) {
    s  += __shfl_xor(s,  off, 32);
    s2 += __shfl_xor(s2, off, 32);
  }
  const float mean = s / (float)N;
  const float var  = s2 / (float)N - mean * mean;
  const float inv  = rsqrtf(var + 1e-5f);
  for (int j = lane; j < N; j += 32) {
    float v = (xr[j] - mean) * inv * g[j] + b[j];
    xr[j] = fmaxf(v, 0.f);
  }
}

// Zero-pad weight pack: W[K x N] -> Wp[Kp x Np]
__global__ void pack_w(const float* __restrict__ W, float* __restrict__ Wp,
                       int K, int N, int Kp, int Np)
{
  const int idx = blockIdx.x * blockDim.x + threadIdx.x;
  if (idx >= Kp * Np) return;
  const int k = idx / Np, n = idx % Np;
  Wp[idx] = (k < K && n < N) ? W[(size_t)k * N + n] : 0.f;
}

// _flow: F[r] = mask ? 0 : (Hb - Ha), mask = (sum(Ha)==0) | (sum(Hb)==0)
__global__ void flow_f32(const float* __restrict__ H, float* __restrict__ F,
                         int rows, int rowOffA, int rowOffB)
{
  const int r = blockIdx.x * 4 + (threadIdx.x >> 5);
  if (r >= rows) return;
  const int lane = threadIdx.x & 31;
  const float* a = H + (size_t)(r + rowOffA) * 50;
  const float* b = H + (size_t)(r + rowOffB) * 50;
  float sa = 0.f, sb = 0.f;
  for (int j = lane; j < 50; j += 32) { sa += a[j]; sb += b[j]; }
#pragma unroll
  for (int off = 16; off > 0; off >>= 1) {
    sa += __shfl_xor(sa, off, 32);
    sb += __shfl_xor(sb, off, 32);
  }
  const bool msk = (sa == 0.f) || (sb == 0.f);
  for (int j = lane; j < 50; j += 32)
    F[(size_t)r * 50 + j] = msk ? 0.f : (b[j] - a[j]);
}

// concat([motion_out(rows,50), H[r+hoff](rows,50)], -1) -> TJ(rows,100)
__global__ void cat_traj(const float* __restrict__ Msrc,
                         const float* __restrict__ H,
                         float* __restrict__ TJ, int rows, int hoff)
{
  const int idx = blockIdx.x * blockDim.x + threadIdx.x;
  if (idx >= rows * 100) return;
  const int r = idx / 100, c = idx % 100;
  TJ[idx] = (c < 50) ? Msrc[(size_t)r * 50 + c]
                     : H[(size_t)(r + hoff) * 50 + (c - 50)];
}

// concat([FP, target=H[t+2], BP], axis=1) -> O(n,39,50)
__global__ void cat_out(const float* __restrict__ FP,
                        const float* __restrict__ H,
                        const float* __restrict__ BP,
                        float* __restrict__ O, int nc)
{
  const int idx = blockIdx.x * blockDim.x + threadIdx.x;
  if (idx >= nc * 39 * 50) return;
  const int c = idx % 50;
  const int rs = idx / 50;
  const int t = rs / 39, s = rs % 39;
  float v;
  if (s < 13)       v = FP[((size_t)t * 13 + s) * 50 + c];
  else if (s < 26)  v = H[((size_t)(t + 2) * 13 + (s - 13)) * 50 + c];
  else              v = BP[((size_t)t * 13 + (s - 26)) * 50 + c];
  O[idx] = v;
}

// ---------------------------------------------------------------------------
// Host side
// ---------------------------------------------------------------------------
static inline int rup(int x, int a) { return (x + a - 1) / a * a; }

struct LDef { int win; int K; int N; };
// d_in flat order: x, then per MLP per layer: W, b, g, beta
static const LDef LAYERS[17] = {
  { 1,   4,  20}, { 5,  20,  50},   // mlp_init
  { 9,   8,  16}, {13,  16,   1},   // mlp_cat_points
  {17,  13,  26}, {21,  26,  13},   // mlp_spatial
  {25,  50, 100}, {29, 100,  50},   // motion
  {33,  50, 100}, {37, 100,  50},   // process_features
  {41, 100, 100}, {45, 100,  50},   // trajectory
  {49,  39,  78}, {53,  78,  39},   // mlp_spatial_2
  {57, 150, 100}, {61, 100,  25}, {65, 25, 3},  // pred
};

static AMap dense_map(int K) { return AMap{1 << 30, 0, K, 1 << 30, 0, 1}; }

static void run_gemm(hipStream_t s, const float* A, const float* Wp,
                     const float* bias, float* Out,
                     int M, int N, int Np, int Ka, int Kp, AMap am)
{
  const int mblk = (M + 16 * GEMM_WAVES - 1) / (16 * GEMM_WAVES);
  const bool dense = (am.ks == 1) && (am.kos == 0) && (am.kc == (1 << 30)) &&
                     ((Ka & 1) == 0);
  if (Np >= 32) {
    dim3 grid((Np + 31) / 32, mblk);
    if (dense) gemm_wmma_f32<true,  true ><<<grid, 128, 0, s>>>(A, Wp, bias, Out, M, N, Np, Ka, Kp, am);
    else       gemm_wmma_f32<true,  false><<<grid, 128, 0, s>>>(A, Wp, bias, Out, M, N, Np, Ka, Kp, am);
  } else {
    dim3 grid(Np / 16, mblk);
    if (dense) gemm_wmma_f32<false, true ><<<grid, 128, 0, s>>>(A, Wp, bias, Out, M, N, Np, Ka, Kp, am);
    else       gemm_wmma_f32<false, false><<<grid, 128, 0, s>>>(A, Wp, bias, Out, M, N, Np, Ka, Kp, am);
  }
}

static void run_ln(hipStream_t s, float* X, const float* g, const float* b,
                   int rows, int N)
{
  lnrelu_f32<<<(rows + 3) / 4, 128, 0, s>>>(X, g, b, rows, N);
}

extern "C" void kernel_launch(void* const* d_in, const int* in_sizes, int n_in,
                              void* d_out, int out_size, void* d_ws, size_t ws_size,
                              hipStream_t stream)
{
  const float* x = (const float*)d_in[0];
  const int T = in_sizes[0] / (13 * 8 * 4);
  const int n = T - 4;
  float* ws = (float*)d_ws;
  float* dout = (float*)d_out;

  // ---- pack weights (zero-padded Kp x Np) at the front of ws ----
  int Kp[17], Np[17];
  size_t woff[17], wtot = 0;
  for (int i = 0; i < 17; ++i) {
    Kp[i] = rup(LAYERS[i].K, 4);
    Np[i] = rup(LAYERS[i].N, 16);
    woff[i] = wtot;
    wtot += (size_t)Kp[i] * Np[i];
  }
  for (int i = 0; i < 17; ++i) {
    const int tot = Kp[i] * Np[i];
    pack_w<<<(tot + 255) / 256, 256, 0, stream>>>(
        (const float*)d_in[LAYERS[i].win], ws + woff[i],
        LAYERS[i].K, LAYERS[i].N, Kp[i], Np[i]);
  }
  const size_t wreg = (wtot + 255) & ~(size_t)255;

  // ---- choose chunk size so the scratch fits in ws ----
  int TC = 256;
  while (TC > 32) {
    size_t need = (wreg + (size_t)(TC + 4) * 22230 + (size_t)TC * 10725) * 4;
    if (need <= ws_size) break;
    TC >>= 1;
  }
  const int Cm = TC + 4;

  // ---- chunk scratch layout (floats; all sizes even -> b64-aligned) ----
  size_t o = wreg;
  auto alloc = [&](size_t fl) { size_t r = o; o += fl; return r; };
  const size_t oB1 = alloc((size_t)Cm * 104 * 20);  // init L0 out
  const size_t oB2 = alloc((size_t)Cm * 104 * 50);  // init L1 out  (T,13,8,50)
  const size_t oB3 = alloc((size_t)Cm * 650 * 16);  // cat L0 out
  const size_t oB4 = alloc((size_t)Cm * 650);       // cat L1 out   (T,13,50)
  const size_t oB5 = alloc((size_t)Cm * 50 * 26);   // spatial L0 out
  const size_t oB6 = alloc((size_t)Cm * 50 * 13);   // spatial L1 out (T,50,13)
  const size_t oB7 = alloc((size_t)Cm * 13 * 100);  // process L0 out
  const size_t oH  = alloc((size_t)Cm * 13 * 50);   // h (C,13,50)
  const size_t oFf = alloc((size_t)TC * 13 * 50);
  const size_t oFb = alloc((size_t)TC * 13 * 50);
  const size_t oFM = alloc((size_t)TC * 13 * 50);
  const size_t oBM = alloc((size_t)TC * 13 * 50);
  const size_t oTJ = alloc((size_t)TC * 13 * 100);
  const size_t oFP = alloc((size_t)TC * 13 * 50);
  const size_t oBP = alloc((size_t)TC * 13 * 50);
  const size_t oO  = alloc((size_t)TC * 39 * 50);
  const size_t oS  = alloc((size_t)TC * 50 * 39);
  const size_t oM1 = alloc((size_t)TC * 13 * 100);
  const size_t oM2 = alloc((size_t)TC * 13 * 25 + 1);

  const float* W[17]; const float* B[17]; const float* G[17]; const float* BE[17];
  for (int i = 0; i < 17; ++i) {
    W[i]  = ws + woff[i];
    B[i]  = (const float*)d_in[LAYERS[i].win + 1];
    G[i]  = (const float*)d_in[LAYERS[i].win + 2];
    BE[i] = (const float*)d_in[LAYERS[i].win + 3];
  }

  for (int c0 = 0; c0 < n; c0 += TC) {
    const int nc = (n - c0 < TC) ? (n - c0) : TC;
    const int C  = nc + 4;                       // h halo
    const float* xc = x + (size_t)c0 * 13 * 8 * 4;

    // ---- mlp_init: (C*104, 4) -> 20 [LN,ReLU] -> 50 ----
    run_gemm(stream, xc, W[0], B[0], ws + oB1, C * 104, 20, Np[0], 4, Kp[0], dense_map(4));
    run_ln(stream, ws + oB1, G[0], BE[0], C * 104, 20);
    run_gemm(stream, ws + oB1, W[1], B[1], ws + oB2, C * 104, 50, Np[1], 20, Kp[1], dense_map(20));

    // ---- mlp_cat_points on transposed (C,13,50,8): rows m=(t*13+j)*50+f, k=p
    run_gemm(stream, ws + oB2, W[2], B[2], ws + oB3, C * 650, 16, Np[2], 8, Kp[2],
             AMap{50, 400, 1, 1 << 30, 0, 50});
    run_ln(stream, ws + oB3, G[2], BE[2], C * 650, 16);
    run_gemm(stream, ws + oB3, W[3], B[3], ws + oB4, C * 650, 1, Np[3], 16, Kp[3], dense_map(16));
    // B4 == (C,13,50) dense

    // ---- mlp_spatial on (C,50,13): rows m=(t*50+f), k=j
    run_gemm(stream, ws + oB4, W[4], B[4], ws + oB5, C * 50, 26, Np[4], 13, Kp[4],
             AMap{50, 650, 1, 1 << 30, 0, 50});
    run_ln(stream, ws + oB5, G[4], BE[4], C * 50, 26);
    run_gemm(stream, ws + oB5, W[5], B[5], ws + oB6, C * 50, 13, Np[5], 26, Kp[5], dense_map(26));
    // B6 == (C,50,13) dense

    // ---- process_features on (C,13,50): rows m=(t*13+j), k=f
    run_gemm(stream, ws + oB6, W[8], B[8], ws + oB7, C * 13, 100, Np[8], 50, Kp[8],
             AMap{13, 650, 1, 1 << 30, 0, 13});
    run_ln(stream, ws + oB7, G[8], BE[8], C * 13, 100);
    run_gemm(stream, ws + oB7, W[9], B[9], ws + oH, C * 13, 50, Np[9], 100, Kp[9], dense_map(100));
    // H == (C,13,50) dense

    // ---- temporal: flow fwd = flow(h[t], h[t+1]); bwd = flow(h[t+3], h[t+4])
    flow_f32<<<(nc * 13 + 3) / 4, 128, 0, stream>>>(ws + oH, ws + oFf, nc * 13, 0, 13);
    flow_f32<<<(nc * 13 + 3) / 4, 128, 0, stream>>>(ws + oH, ws + oFb, nc * 13, 39, 52);

    // ---- motion MLP (50 -> 100 [LN,ReLU] -> 50) on both flows ----
    run_gemm(stream, ws + oFf, W[6], B[6], ws + oM1, nc * 13, 100, Np[6], 50, Kp[6], dense_map(50));
    run_ln(stream, ws + oM1, G[6], BE[6], nc * 13, 100);
    run_gemm(stream, ws + oM1, W[7], B[7], ws + oFM, nc * 13, 50, Np[7], 100, Kp[7], dense_map(100));
    run_gemm(stream, ws + oFb, W[6], B[6], ws + oM1, nc * 13, 100, Np[6], 50, Kp[6], dense_map(50));
    run_ln(stream, ws + oM1, G[6], BE[6], nc * 13, 100);
    run_gemm(stream, ws + oM1, W[7], B[7], ws + oBM, nc * 13, 50, Np[7], 100, Kp[7], dense_map(100));

    // ---- trajectory: concat([motion, h[t+1|t+3]], -1) -> 100 -> 100 [LN,ReLU] -> 50
    {
      const int tot = nc * 13 * 100;
      cat_traj<<<(tot + 255) / 256, 256, 0, stream>>>(ws + oFM, ws + oH, ws + oTJ, nc * 13, 13);
      run_gemm(stream, ws + oTJ, W[10], B[10], ws + oM1, nc * 13, 100, Np[10], 100, Kp[10], dense_map(100));
      run_ln(stream, ws + oM1, G[10], BE[10], nc * 13, 100);
      run_gemm(stream, ws + oM1, W[11], B[11], ws + oFP, nc * 13, 50, Np[11], 100, Kp[11], dense_map(100));

      cat_traj<<<(tot + 255) / 256, 256, 0, stream>>>(ws + oBM, ws + oH, ws + oTJ, nc * 13, 39);
      run_gemm(stream, ws + oTJ, W[10], B[10], ws + oM1, nc * 13, 100, Np[10], 100, Kp[10], dense_map(100));
      run_ln(stream, ws + oM1, G[10], BE[10], nc * 13, 100);
      run_gemm(stream, ws + oM1, W[11], B[11], ws + oBP, nc * 13, 50, Np[11], 100, Kp[11], dense_map(100));
    }

    // ---- out = concat([fp, target, bp], joints) -> (nc,39,50) ----
    {
      const int tot = nc * 39 * 50;
      cat_out<<<(tot + 255) / 256, 256, 0, stream>>>(ws + oFP, ws + oH, ws + oBP, ws + oO, nc);
    }

    // ---- mlp_spatial_2 on (nc,50,39): rows m=(t*50+f), k=r
    run_gemm(stream, ws + oO, W[12], B[12], ws + oM1, nc * 50, 78, Np[12], 39, Kp[12],
             AMap{50, 1950, 1, 1 << 30, 0, 50});
    run_ln(stream, ws + oM1, G[12], BE[12], nc * 50, 78);
    run_gemm(stream, ws + oM1, W[13], B[13], ws + oS, nc * 50, 39, Np[13], 78, Kp[13], dense_map(78));
    // S == (nc,50,39) dense

    // ---- pred: input gather (nc,13,150): addr=(m/13)*1950+(m%13)+(k/50)*13+(k%50)*39
    run_gemm(stream, ws + oS, W[14], B[14], ws + oM1, nc * 13, 100, Np[14], 150, Kp[14],
             AMap{13, 1950, 1, 50, 13, 39});
    run_ln(stream, ws + oM1, G[14], BE[14], nc * 13, 100);
    run_gemm(stream, ws + oM1, W[15], B[15], ws + oM2, nc * 13, 25, Np[15], 100, Kp[15], dense_map(100));
    run_ln(stream, ws + oM2, G[15], BE[15], nc * 13, 25);
    run_gemm(stream, ws + oM2, W[16], B[16], dout + (size_t)c0 * 39, nc * 13, 3, Np[16], 25, Kp[16],
             dense_map(25));
  }
}